// GineEncoder_11605001633946
// MI455X (gfx1250) — compile-verified
//
#include <hip/hip_runtime.h>

typedef __attribute__((ext_vector_type(2))) float v2f;
typedef __attribute__((ext_vector_type(8))) float v8f;

#define NNODES 50000
#define NEDGES 800000
#define FNODE 32
#define FEDGE 16
#define HID 96
#define NLAYERS 3
#define NGRAPHS 50
#define OUTCOLS (NLAYERS * HID)   // 288
#define BN_EPS 1e-5f

static __device__ __forceinline__ v8f wmma_f32(v2f a, v2f b, v8f c) {
  // D = A(16x4 f32) * B(4x16 f32) + C(16x16 f32)
  return __builtin_amdgcn_wmma_f32_16x16x4_f32(false, a, false, b, (short)0, c,
                                               false, false);
}

// ---------------------------------------------------------------------------
// Fused edge kernel: emb = edge_attr @ We + be  (WMMA, K=16)
// then msg = relu(h[src] + emb) scatter-added into agg[dst].
// One wave per tile of 16 edges; D = node feature width (32 or 96),
// compile-time so strides strength-reduce and offsets stay 32-bit (SADDR form).
// ---------------------------------------------------------------------------
template <int D>
__global__ __launch_bounds__(256) void edge_msg_scatter(
    const float* __restrict__ h,
    const float* __restrict__ edge_attr,          // [E x 16]
    const int* __restrict__ src, const int* __restrict__ dst,
    const float* __restrict__ We,                 // [16 x D]
    const float* __restrict__ be,                 // [D]
    float* __restrict__ agg)                      // [N x D]
{
  const int lane = threadIdx.x & 31;
  const int wave = threadIdx.x >> 5;
  const int tile = blockIdx.x * (blockDim.x >> 5) + wave;
  const int ntiles = NEDGES / 16;
  if (tile >= ntiles) return;                 // wave-uniform

  const int e0   = tile * 16;
  const int half = lane >> 4;                 // 0 or 1
  const int l15  = lane & 15;

  // Preload A fragments for K=0..15 (edge_attr rows = edges of this tile).
  v2f afr[4];
#pragma unroll
  for (int kk = 0; kk < 4; ++kk) {
    const int aoff = (e0 + l15) * FEDGE + kk * 4 + half * 2;  // 32-bit
    afr[kk].x = edge_attr[aoff];
    afr[kk].y = edge_attr[aoff + 1];
  }
  // Preload src/dst row bases (in elements) for this lane-half's 8 edges.
  int sbase[8], dbase[8];
#pragma unroll
  for (int r = 0; r < 8; ++r) {
    const int e = e0 + r + half * 8;
    sbase[r] = src[e] * D;
    dbase[r] = dst[e] * D;
  }

#pragma unroll
  for (int tn = 0; tn < D / 16; ++tn) {
    const int n = tn * 16 + l15;
    const float bias = be[n];
    v8f acc;
#pragma unroll
    for (int r = 0; r < 8; ++r) acc[r] = bias;
#pragma unroll
    for (int kk = 0; kk < 4; ++kk) {
      const int ka = kk * 4 + half * 2;
      v2f b;
      b.x = We[ka * D + n];
      b.y = We[(ka + 1) * D + n];
      acc = wmma_f32(afr[kk], b, acc);
    }
#pragma unroll
    for (int r = 0; r < 8; ++r) {
      float v = h[sbase[r] + n] + acc[r];
      v = v > 0.0f ? v : 0.0f;
      atomicAdd(&agg[dbase[r] + n], v);
    }
  }
}

// ---------------------------------------------------------------------------
// GEMM1: y = (h + agg) @ W + bias, plus per-feature sum/sumsq for BN.
// One wave per 16x16 output tile. K compile-time (32 or 96).
// ---------------------------------------------------------------------------
template <int K>
__global__ __launch_bounds__(256) void gemm_node1(
    const float* __restrict__ h, const float* __restrict__ agg,
    const float* __restrict__ W,      // [K x HID]
    const float* __restrict__ bias,   // [HID]
    float* __restrict__ y,            // [N x HID]
    double* __restrict__ bn_sum, double* __restrict__ bn_sqs)
{
  const int lane = threadIdx.x & 31;
  const int wave = threadIdx.x >> 5;
  const int NT = HID / 16;                    // 6
  const int MT = NNODES / 16;                 // 3125
  const int tile = blockIdx.x * (blockDim.x >> 5) + wave;
  if (tile >= MT * NT) return;                // wave-uniform

  const int tm = tile / NT, tn = tile % NT;
  const int half = lane >> 4, l15 = lane & 15;
  const int n = tn * 16 + l15;
  const int row = tm * 16 + l15;

  v8f acc;
  const float b0 = bias[n];
#pragma unroll
  for (int r = 0; r < 8; ++r) acc[r] = b0;

#pragma unroll
  for (int k = 0; k < K; k += 4) {
    const int ka = k + half * 2;
    const int off = row * K + ka;             // 32-bit offsets -> SADDR form
    v2f a, b;
    a.x = h[off] + agg[off];
    a.y = h[off + 1] + agg[off + 1];
    b.x = W[ka * HID + n];
    b.y = W[(ka + 1) * HID + n];
    acc = wmma_f32(a, b, acc);
  }

  float s = 0.f, s2 = 0.f;
#pragma unroll
  for (int r = 0; r < 8; ++r) {
    const int rr = tm * 16 + r + half * 8;
    const float v = acc[r];
    y[rr * HID + n] = v;
    s += v;
    s2 += v * v;
  }
  atomicAdd(&bn_sum[n], (double)s);
  atomicAdd(&bn_sqs[n], (double)s2);
}

// ---------------------------------------------------------------------------
// GEMM2: y2 = relu(bnA*y1 + bnC) @ W + bias, plus BN2 sums. K = HID.
// ---------------------------------------------------------------------------
__global__ __launch_bounds__(256) void gemm_node2(
    const float* __restrict__ yin,
    const float* __restrict__ bnA, const float* __restrict__ bnC,
    const float* __restrict__ W, const float* __restrict__ bias,
    float* __restrict__ yout,
    double* __restrict__ bn_sum, double* __restrict__ bn_sqs)
{
  const int lane = threadIdx.x & 31;
  const int wave = threadIdx.x >> 5;
  const int NT = HID / 16;
  const int MT = NNODES / 16;
  const int tile = blockIdx.x * (blockDim.x >> 5) + wave;
  if (tile >= MT * NT) return;

  const int tm = tile / NT, tn = tile % NT;
  const int half = lane >> 4, l15 = lane & 15;
  const int n = tn * 16 + l15;
  const int row = tm * 16 + l15;

  v8f acc;
  const float b0 = bias[n];
#pragma unroll
  for (int r = 0; r < 8; ++r) acc[r] = b0;

#pragma unroll
  for (int k = 0; k < HID; k += 4) {
    const int ka = k + half * 2;
    const int off = row * HID + ka;
    v2f a, b;
    float x0 = fmaf(bnA[ka], yin[off], bnC[ka]);
    float x1 = fmaf(bnA[ka + 1], yin[off + 1], bnC[ka + 1]);
    a.x = x0 > 0.f ? x0 : 0.f;
    a.y = x1 > 0.f ? x1 : 0.f;
    b.x = W[ka * HID + n];
    b.y = W[(ka + 1) * HID + n];
    acc = wmma_f32(a, b, acc);
  }

  float s = 0.f, s2 = 0.f;
#pragma unroll
  for (int r = 0; r < 8; ++r) {
    const int rr = tm * 16 + r + half * 8;
    const float v = acc[r];
    yout[rr * HID + n] = v;
    s += v;
    s2 += v * v;
  }
  atomicAdd(&bn_sum[n], (double)s);
  atomicAdd(&bn_sqs[n], (double)s2);
}

// Fold BN into per-feature affine: a = g*rsqrt(var+eps), c = bt - a*mu.
__global__ void bn_finalize(const double* __restrict__ sum,
                            const double* __restrict__ sqs,
                            const float* __restrict__ g,
                            const float* __restrict__ bt,
                            float* __restrict__ A, float* __restrict__ C)
{
  const int f = threadIdx.x;
  if (f < HID) {
    const double inv_n = 1.0 / (double)NNODES;
    const float mu = (float)(sum[f] * inv_n);
    const float var = (float)(sqs[f] * inv_n) - mu * mu;
    const float a = g[f] * rsqrtf(var + BN_EPS);
    A[f] = a;
    C[f] = bt[f] - a * mu;
  }
}

// h_next = relu(bnA*y + bnC); pooled segment-sum into out[:, layer*HID ...].
// float4 per thread: 24 threads per node.
__global__ __launch_bounds__(256) void apply_pool(
    const float* __restrict__ y,
    const float* __restrict__ bnA, const float* __restrict__ bnC,
    const int* __restrict__ batch,
    float* __restrict__ hout, float* __restrict__ out, int layer)
{
  const int q = blockIdx.x * blockDim.x + threadIdx.x;   // quad index
  if (q >= NNODES * (HID / 4)) return;
  const int i = q / (HID / 4);
  const int f = (q - i * (HID / 4)) * 4;
  const int off = i * HID + f;

  const float4 yv = *(const float4*)(y + off);
  float4 v;
  v.x = fmaf(bnA[f + 0], yv.x, bnC[f + 0]);
  v.y = fmaf(bnA[f + 1], yv.y, bnC[f + 1]);
  v.z = fmaf(bnA[f + 2], yv.z, bnC[f + 2]);
  v.w = fmaf(bnA[f + 3], yv.w, bnC[f + 3]);
  v.x = v.x > 0.f ? v.x : 0.f;
  v.y = v.y > 0.f ? v.y : 0.f;
  v.z = v.z > 0.f ? v.z : 0.f;
  v.w = v.w > 0.f ? v.w : 0.f;
  *(float4*)(hout + off) = v;

  const int obase = batch[i] * OUTCOLS + layer * HID + f;
  atomicAdd(&out[obase + 0], v.x);
  atomicAdd(&out[obase + 1], v.y);
  atomicAdd(&out[obase + 2], v.z);
  atomicAdd(&out[obase + 3], v.w);
}

// ---------------------------------------------------------------------------
extern "C" void kernel_launch(void* const* d_in, const int* in_sizes, int n_in,
                              void* d_out, int out_size, void* d_ws, size_t ws_size,
                              hipStream_t stream) {
  (void)in_sizes; (void)n_in; (void)out_size; (void)ws_size;

  const float* x         = (const float*)d_in[0];
  const int*   edge_idx  = (const int*)d_in[1];
  const int*   batch     = (const int*)d_in[2];
  const float* edge_attr = (const float*)d_in[3];
  // params flattened: per layer {We,be,W1,b1,g1,bt1,W2,b2,g2,bt2}
  auto prm = [&](int layer, int j) -> const float* {
    return (const float*)d_in[4 + layer * 10 + j];
  };

  const int* src = edge_idx;
  const int* dst = edge_idx + NEDGES;

  // Workspace carve (base from hipMalloc is 256B aligned).
  float* agg  = (float*)d_ws;                       // [N x HID] (also GEMM2 out)
  float* ybuf = agg  + (size_t)NNODES * HID;        // [N x HID]
  float* hbuf = ybuf + (size_t)NNODES * HID;        // [N x HID]
  float* bnA  = hbuf + (size_t)NNODES * HID;        // [HID]
  float* bnC  = bnA + HID;                          // [HID]
  double* bn_sum = (double*)(bnC + HID);            // [HID] (8B-aligned offset)
  double* bn_sqs = bn_sum + HID;                    // [HID]

  hipMemsetAsync(d_out, 0, sizeof(float) * NGRAPHS * OUTCOLS, stream);

  const int etiles = NEDGES / 16;                   // 50000
  const int egrid  = (etiles + 7) / 8;
  const int ntiles = (NNODES / 16) * (HID / 16);    // 18750
  const int ngrid  = (ntiles + 7) / 8;

  const float* h = x;
  for (int layer = 0; layer < NLAYERS; ++layer) {
    const int d = (layer == 0) ? FNODE : HID;
    hipMemsetAsync(agg, 0, sizeof(float) * (size_t)NNODES * d, stream);

    if (layer == 0)
      edge_msg_scatter<FNODE><<<egrid, 256, 0, stream>>>(
          h, edge_attr, src, dst, prm(layer, 0), prm(layer, 1), agg);
    else
      edge_msg_scatter<HID><<<egrid, 256, 0, stream>>>(
          h, edge_attr, src, dst, prm(layer, 0), prm(layer, 1), agg);

    hipMemsetAsync(bn_sum, 0, sizeof(double) * 2 * HID, stream);
    if (layer == 0)
      gemm_node1<FNODE><<<ngrid, 256, 0, stream>>>(
          h, agg, prm(layer, 2), prm(layer, 3), ybuf, bn_sum, bn_sqs);
    else
      gemm_node1<HID><<<ngrid, 256, 0, stream>>>(
          h, agg, prm(layer, 2), prm(layer, 3), ybuf, bn_sum, bn_sqs);
    bn_finalize<<<1, HID, 0, stream>>>(bn_sum, bn_sqs, prm(layer, 4),
                                       prm(layer, 5), bnA, bnC);

    hipMemsetAsync(bn_sum, 0, sizeof(double) * 2 * HID, stream);
    gemm_node2<<<ngrid, 256, 0, stream>>>(
        ybuf, bnA, bnC, prm(layer, 6), prm(layer, 7), agg, bn_sum, bn_sqs);
    bn_finalize<<<1, HID, 0, stream>>>(bn_sum, bn_sqs, prm(layer, 8),
                                       prm(layer, 9), bnA, bnC);

    apply_pool<<<(NNODES * (HID / 4) + 255) / 256, 256, 0, stream>>>(
        agg, bnA, bnC, batch, hbuf, (float*)d_out, layer);

    h = hbuf;
  }
}